// SilkNNUE_76742475645269
// MI455X (gfx1250) — compile-verified
//
#include <hip/hip_runtime.h>

// CDNA5 / gfx1250, wave32. fp32 end-to-end (matches reference numerics).
// Per wave: 16 sample rows. GEMMs via V_WMMA_F32_16X16X4_F32.

typedef __attribute__((ext_vector_type(2))) float v2f;
typedef __attribute__((ext_vector_type(4))) float v4f;
typedef __attribute__((ext_vector_type(8))) float v8f;

#define ROWSTRIDE 132   // 128 + 4 pad floats: avoids LDS bank conflicts, keeps 16B align

__global__ __launch_bounds__(256) void nnue_kernel(
    const int* __restrict__ x,      // (nrows, 32) int32
    const float* __restrict__ emb,  // (14848, 128)
    const float* __restrict__ w2,   // (32, 128)
    const float* __restrict__ b2,   // (32,)
    const float* __restrict__ w3,   // (32, 64)
    const float* __restrict__ b3,   // (32,)
    const float* __restrict__ w4,   // (1, 64)
    float* __restrict__ out,        // (nrows,)
    int nrows)
{
  __shared__ float sh[8 * 16 * ROWSTRIDE];   // 67,584 B: per-wave 16x132 activation tile

  const int lane = threadIdx.x & 31;
  const int wave = threadIdx.x >> 5;
  const int row0 = blockIdx.x * 128 + wave * 16;
  float* shw = sh + wave * 16 * ROWSTRIDE;

  // ---------------- Phase 1: embedding gather-sum + ReLU -> LDS h[16][128] ---
  // Whole wave loads one 512B embedding row per step (32 lanes x float4), fully
  // coalesced; table (7.6MB) is L2-resident, so this is the L2-bound hot loop.
  for (int r = 0; r < 16; ++r) {
    const int* xr = x + (size_t)(row0 + r) * 32;
    v4f acc = {0.f, 0.f, 0.f, 0.f};
#pragma unroll 4
    for (int j = 0; j < 29; ++j) {          // reference uses x[..., :29]
      int idx = xr[j];                      // wave-uniform index
      const v4f* e = (const v4f*)(emb + (size_t)idx * 128 + lane * 4);
      acc += *e;
    }
    v4f h;
    h.x = fmaxf(acc.x, 0.f);
    h.y = fmaxf(acc.y, 0.f);
    h.z = fmaxf(acc.z, 0.f);
    h.w = fmaxf(acc.w, 0.f);
    *(v4f*)(shw + r * ROWSTRIDE + lane * 4) = h;
  }

  // WMMA fp32 fragment coordinates (16x4 A layout / 4x16 B layout):
  // lane l -> row/col = l%16, K pair offset = 2*(l/16).
  const int mrow = lane & 15;
  const int koff = (lane >> 4) << 1;
  const float* aBase = shw + mrow * ROWSTRIDE;

  // ---------------- Phase 2: layer 2 GEMM (K=128, N=32) with WMMA -----------
  const float* b2t0 = w2 + (size_t)mrow * 128;         // B tile0: n = mrow
  const float* b2t1 = w2 + (size_t)(mrow + 16) * 128;  // B tile1: n = mrow+16
  v8f c0 = {}; v8f c1 = {};
#pragma unroll
  for (int k0 = 0; k0 < 128; k0 += 4) {
    v2f a  = *(const v2f*)(aBase + k0 + koff);
    v2f bb0 = *(const v2f*)(b2t0 + k0 + koff);
    v2f bb1 = *(const v2f*)(b2t1 + k0 + koff);
    c0 = __builtin_amdgcn_wmma_f32_16x16x4_f32(false, a, false, bb0, (short)0, c0, false, false);
    c1 = __builtin_amdgcn_wmma_f32_16x16x4_f32(false, a, false, bb1, (short)0, c1, false, false);
  }

  // Bias + CReLU (concat(h,-h) relu) -> LDS a3[16][64] (reuse same tile).
  // D layout: vgpr v, lanes 0-15 -> M=v, N=lane; lanes 16-31 -> M=v+8, N=lane-16.
  {
    const float bias0 = b2[mrow];
    const float bias1 = b2[mrow + 16];
    const int mbase = (lane >> 4) * 8;
#pragma unroll
    for (int v = 0; v < 8; ++v) {
      float h0 = c0[v] + bias0;
      float h1 = c1[v] + bias1;
      float* rowp = shw + (mbase + v) * ROWSTRIDE;
      rowp[mrow]      = fmaxf(h0, 0.f);
      rowp[mrow + 16] = fmaxf(h1, 0.f);
      rowp[mrow + 32] = fmaxf(-h0, 0.f);
      rowp[mrow + 48] = fmaxf(-h1, 0.f);
    }
  }

  // ---------------- Phase 3: layer 3 GEMM (K=64, N=32) with WMMA ------------
  const float* b3t0 = w3 + (size_t)mrow * 64;
  const float* b3t1 = w3 + (size_t)(mrow + 16) * 64;
  v8f d0 = {}; v8f d1 = {};
#pragma unroll
  for (int k0 = 0; k0 < 64; k0 += 4) {
    v2f a  = *(const v2f*)(aBase + k0 + koff);
    v2f bb0 = *(const v2f*)(b3t0 + k0 + koff);
    v2f bb1 = *(const v2f*)(b3t1 + k0 + koff);
    d0 = __builtin_amdgcn_wmma_f32_16x16x4_f32(false, a, false, bb0, (short)0, d0, false, false);
    d1 = __builtin_amdgcn_wmma_f32_16x16x4_f32(false, a, false, bb1, (short)0, d1, false, false);
  }

  // ---------------- Phase 4: bias + CReLU + w4 dot, cross-lane reduce -------
  {
    const float bias0 = b3[mrow];
    const float bias1 = b3[mrow + 16];
    const float w4p0 = w4[mrow];        // weight for relu(+h), tile0 (n)
    const float w4p1 = w4[mrow + 16];   // tile1 (n+16)
    const float w4n0 = w4[mrow + 32];   // weight for relu(-h), tile0
    const float w4n1 = w4[mrow + 48];   // tile1
#pragma unroll
    for (int v = 0; v < 8; ++v) {
      float h0 = d0[v] + bias0;
      float h1 = d1[v] + bias1;
      float p = fmaxf(h0, 0.f) * w4p0 + fmaxf(-h0, 0.f) * w4n0
              + fmaxf(h1, 0.f) * w4p1 + fmaxf(-h1, 0.f) * w4n1;
      // Reduce over the 16 lanes of each half (masks 1,2,4,8 stay within halves).
      p += __shfl_xor(p, 1);
      p += __shfl_xor(p, 2);
      p += __shfl_xor(p, 4);
      p += __shfl_xor(p, 8);
      if (lane == 0 && row0 + v < nrows)       out[row0 + v]     = p;   // M = v
      if (lane == 16 && row0 + 8 + v < nrows)  out[row0 + 8 + v] = p;   // M = v+8
    }
  }
}

extern "C" void kernel_launch(void* const* d_in, const int* in_sizes, int n_in,
                              void* d_out, int out_size, void* d_ws, size_t ws_size,
                              hipStream_t stream) {
  const int*   x   = (const int*)d_in[0];
  const float* emb = (const float*)d_in[1];
  const float* w2  = (const float*)d_in[2];
  const float* b2  = (const float*)d_in[3];
  const float* w3  = (const float*)d_in[4];
  const float* b3  = (const float*)d_in[5];
  const float* w4  = (const float*)d_in[6];
  float* out = (float*)d_out;

  const int nrows = in_sizes[0] / 32;      // 65536
  const int blocks = nrows / 128;          // 512 (nrows is an exact multiple of 128)
  nnue_kernel<<<blocks, 256, 0, stream>>>(x, emb, w2, b2, w3, b3, w4, out, nrows);
}